// GATLayerAdj_82197084111206
// MI455X (gfx1250) — compile-verified
//
#include <hip/hip_runtime.h>
#include <math.h>

typedef __attribute__((ext_vector_type(2))) float v2f;
typedef __attribute__((ext_vector_type(8))) float v8f;

#define NVERT 1024
#define DI 64
#define DO 64
#define NEGV (-1e16f)

// ---------------------------------------------------------------------------
// Kernel 1: ysrc = x @ Wf[:, :di]^T + bf ; ytgt = x @ Wf[:, di:]^T
// One wave (32 threads) per 16x16 output tile, V_WMMA_F32_16X16X4_F32, K=64.
// A layout (16x4 f32): lanes 0-15 hold M=0..15 with K={k0,k0+1}; lanes 16-31
// hold the same rows with K={k0+2,k0+3}. B (4x16) mirrors with N across lanes.
// C layout: VGPR r holds M=r (lanes 0-15) / M=r+8 (lanes 16-31), N=lane&15.
// ---------------------------------------------------------------------------
__global__ __launch_bounds__(32)
void proj_wmma(const float* __restrict__ x, const float* __restrict__ Wf,
               const float* __restrict__ bf,
               float* __restrict__ ysrc, float* __restrict__ ytgt) {
  const int lane = threadIdx.x;        // block = exactly one wave32
  const int half = lane >> 4;          // 0 or 1
  const int l16  = lane & 15;
  const int m0   = blockIdx.x * 16;    // row tile in [0,1024)
  const int n0   = blockIdx.y * 16;    // col tile in [0,64)
  const int which = blockIdx.z;        // 0 = src half, 1 = tgt half
  const int koff  = which * DI;        // column offset into Wf [do, 2*di]

  v8f c = {};
#pragma unroll
  for (int k0 = 0; k0 < DI; k0 += 4) {
    const int ka = k0 + 2 * half;
    v2f a, b;
    // A[m,k] = x[m0+l16, ka..ka+1]
    a.x = x[(m0 + l16) * DI + ka];
    a.y = x[(m0 + l16) * DI + ka + 1];
    // B[k,n] = Wf[n0+l16, koff + k]  (since y = x @ W^T)
    b.x = Wf[(n0 + l16) * (2 * DI) + koff + ka];
    b.y = Wf[(n0 + l16) * (2 * DI) + koff + ka + 1];
    c = __builtin_amdgcn_wmma_f32_16x16x4_f32(
        /*neg_a=*/false, a, /*neg_b=*/false, b,
        /*c_mod=*/(short)0, c, /*reuse_a=*/false, /*reuse_b=*/false);
  }

  float* out = which ? ytgt : ysrc;
  const float bias = which ? 0.0f : bf[n0 + l16];  // fold bf into ysrc
#pragma unroll
  for (int r = 0; r < 8; ++r) {
    const int m = m0 + r + 8 * half;
    out[m * DO + n0 + l16] = c[r] + bias;
  }
}

// ---------------------------------------------------------------------------
// Kernel 2: per-node attention logit pieces (tiny GEMV).
// asrc[j] = x[j]@Ww[0,:di] ; atgt[i] = x[i]@Ww[0,di:] + bw  (bw folded here)
// ---------------------------------------------------------------------------
__global__ __launch_bounds__(256)
void attn_gemv(const float* __restrict__ x, const float* __restrict__ Ww,
               const float* __restrict__ bw,
               float* __restrict__ asrc, float* __restrict__ atgt) {
  const int row = blockIdx.x * 256 + threadIdx.x;
  float s = 0.f, t = 0.f;
#pragma unroll 8
  for (int k = 0; k < DI; ++k) {
    const float xv = x[row * DI + k];   // Ww reads are wave-uniform -> SMEM
    s = fmaf(xv, Ww[k], s);
    t = fmaf(xv, Ww[DI + k], t);
  }
  asrc[row] = s;
  atgt[row] = t + bw[0];
}

// ---------------------------------------------------------------------------
// Kernel 3: fused masked softmax + weighted ReLU einsum, row-stationary.
// One block (8 waves) per destination row i; att probabilities live in LDS.
//   phase 1: logits + block max          phase 2: exp + block sum
//   phase 3: o[i,d] = inv * sum_j p[j]*relu(ysrc[j,d]+ytgt[i,d])
// Thread t -> d = t&63, j-group g = t>>6 (uniform per wave => p[j] is an
// LDS same-address broadcast; ysrc loads are 128B coalesced per wave).
// ---------------------------------------------------------------------------
__global__ __launch_bounds__(256)
void gat_row(const float* __restrict__ ysrc, const float* __restrict__ ytgt,
             const float* __restrict__ asrc, const float* __restrict__ atgt,
             const int* __restrict__ adj, float* __restrict__ o) {
  __shared__ float p[NVERT];
  __shared__ float redm[8];
  __shared__ float reds[8];
  __shared__ float accbuf[4][DO];

  const int tid = threadIdx.x;
  const int i   = blockIdx.x;
  const float ai = atgt[i];             // bw already folded in

  // phase 1: masked logits into LDS, local max
  float lmax = -INFINITY;
  for (int j = tid; j < NVERT; j += 256) {
    const float l = (adj[i * NVERT + j] > 0) ? (asrc[j] + ai) : NEGV;
    p[j] = l;
    lmax = fmaxf(lmax, l);
  }
#pragma unroll
  for (int off = 16; off; off >>= 1) lmax = fmaxf(lmax, __shfl_xor(lmax, off, 32));
  const int wid = tid >> 5;
  if ((tid & 31) == 0) redm[wid] = lmax;
  __syncthreads();
  float m = redm[0];
#pragma unroll
  for (int w = 1; w < 8; ++w) m = fmaxf(m, redm[w]);

  // phase 2: exponentiate in place, local sum
  // (all-masked row: l - m == 0 exactly -> uniform 1/N like jax softmax)
  float lsum = 0.f;
  for (int j = tid; j < NVERT; j += 256) {
    const float e = __expf(p[j] - m);
    p[j] = e;
    lsum += e;
  }
#pragma unroll
  for (int off = 16; off; off >>= 1) lsum += __shfl_xor(lsum, off, 32);
  if ((tid & 31) == 0) reds[wid] = lsum;
  __syncthreads();                      // also publishes all p[j] writes
  float ssum = 0.f;
#pragma unroll
  for (int w = 0; w < 8; ++w) ssum += reds[w];
  const float inv = 1.0f / ssum;

  // phase 3: weighted ReLU accumulation, 4-way split over j
  const int d = tid & 63;
  const int g = tid >> 6;               // 0..3, uniform within a wave
  const float vt = ytgt[i * DO + d];
  float acc = 0.f;
#pragma unroll 8
  for (int j = g; j < NVERT; j += 4) {
    const float val = fmaxf(ysrc[j * DO + d] + vt, 0.f);
    acc = fmaf(p[j], val, acc);
  }
  accbuf[g][d] = acc;
  __syncthreads();
  if (tid < DO) {
    const float r = (accbuf[0][tid] + accbuf[1][tid]) +
                    (accbuf[2][tid] + accbuf[3][tid]);
    o[i * DO + tid] = r * inv;
  }
}

// ---------------------------------------------------------------------------
extern "C" void kernel_launch(void* const* d_in, const int* in_sizes, int n_in,
                              void* d_out, int out_size, void* d_ws, size_t ws_size,
                              hipStream_t stream) {
  const float* x   = (const float*)d_in[0];   // [1024, 64]
  const int*   adj = (const int*)  d_in[1];   // [1024, 1024]
  const float* Wf  = (const float*)d_in[2];   // [64, 128]
  const float* bf  = (const float*)d_in[3];   // [64]
  const float* Ww  = (const float*)d_in[4];   // [1, 128]
  const float* bw  = (const float*)d_in[5];   // [1]
  float* out = (float*)d_out;                 // [1024, 64]

  float* ws   = (float*)d_ws;
  float* ysrc = ws;                            // 1024*64
  float* ytgt = ws + NVERT * DO;               // 1024*64
  float* as   = ws + 2 * NVERT * DO;           // 1024
  float* at   = as + NVERT;                    // 1024

  proj_wmma<<<dim3(NVERT / 16, DO / 16, 2), 32, 0, stream>>>(x, Wf, bf, ysrc, ytgt);
  attn_gemv<<<dim3(NVERT / 256), 256, 0, stream>>>(x, Ww, bw, as, at);
  gat_row<<<dim3(NVERT), 256, 0, stream>>>(ysrc, ytgt, as, at, adj, out);
}